// GaussianMixtureModel_81398220194342
// MI455X (gfx1250) — compile-verified
//
#include <hip/hip_runtime.h>
#include <hip/hip_bf16.h>
#include <math.h>

typedef float v2f __attribute__((ext_vector_type(2)));
typedef float v8f __attribute__((ext_vector_type(8)));

#define P16 16
#define G8  8
// ws layout (floats): Sinv[8*256] | bvec[8*16] | Kc[8] | tileSums[nTiles]
#define WS_SINV  0
#define WS_BVEC  2048
#define WS_KC    2176
#define WS_TILES 2184

// -8 * log(2*pi)  (lead term for P=16)
#define LEAD_F (-14.70301653127476f)

// ---------------------------------------------------------------------------
// Prep kernel: 1 wave; lane g (g<8) does Cholesky of sigmas[g], builds
// Sinv, b = Sinv*mu, K_g = -0.5*mu'Sinv mu - 0.5*logdet + log softmax(w)_g
// ---------------------------------------------------------------------------
__global__ void gmm_prep_kernel(const float* __restrict__ mus,
                                const float* __restrict__ sigmas,
                                const float* __restrict__ w,
                                float* __restrict__ ws) {
    int g = threadIdx.x;
    if (g >= G8) return;
    float* Sinv = ws + WS_SINV;
    float* bvec = ws + WS_BVEC;
    float* Kc   = ws + WS_KC;

    float L[P16][P16];
    for (int p = 0; p < P16; ++p)
        for (int q = 0; q < P16; ++q)
            L[p][q] = sigmas[g * 256 + p * 16 + q];

    // Cholesky (SPD by construction: 0.25*I + A A^T)
    float logdet = 0.f;
    for (int j = 0; j < P16; ++j) {
        float s = L[j][j];
        for (int k = 0; k < j; ++k) s -= L[j][k] * L[j][k];
        float d = sqrtf(s);
        logdet += 2.f * __logf(d);
        L[j][j] = d;
        float inv = 1.f / d;
        for (int i = j + 1; i < P16; ++i) {
            float s2 = L[i][j];
            for (int k = 0; k < j; ++k) s2 -= L[i][k] * L[j][k];
            L[i][j] = s2 * inv;
        }
    }
    // Invert lower-triangular L
    float Li[P16][P16];
    for (int p = 0; p < P16; ++p)
        for (int q = 0; q < P16; ++q) Li[p][q] = 0.f;
    for (int j = 0; j < P16; ++j) {
        Li[j][j] = 1.f / L[j][j];
        for (int i = j + 1; i < P16; ++i) {
            float s = 0.f;
            for (int k = j; k < i; ++k) s += L[i][k] * Li[k][j];
            Li[i][j] = -s / L[i][i];
        }
    }
    // Sinv = Li^T Li ; b = Sinv*mu ; c = mu^T Sinv mu
    float mu[P16];
    for (int p = 0; p < P16; ++p) mu[p] = mus[g * 16 + p];
    float c = 0.f;
    for (int p = 0; p < P16; ++p) {
        float bp = 0.f;
        for (int q = 0; q < P16; ++q) {
            int k0 = p > q ? p : q;
            float s = 0.f;
            for (int k = k0; k < P16; ++k) s += Li[k][p] * Li[k][q];
            Sinv[g * 256 + p * 16 + q] = s;
            bp += s * mu[q];
        }
        bvec[g * 16 + p] = bp;
        c += bp * mu[p];
    }
    // log softmax(w)
    float wm = w[0];
    for (int j = 1; j < G8; ++j) wm = fmaxf(wm, w[j]);
    float se = 0.f;
    for (int j = 0; j < G8; ++j) se += __expf(w[j] - wm);
    float logW = w[g] - wm - __logf(se);
    Kc[g] = -0.5f * c - 0.5f * logdet + logW;
}

// ---------------------------------------------------------------------------
// Main kernel: each wave owns 16-point tiles. Per tile:
//   Box-Cox transform + jacobian (exp/log), stage Xt in LDS,
//   re-read in WMMA A-layout and C-layout, then for each of 8 groups:
//   T = Y @ Sinv_g via 4x v_wmma_f32_16x16x4_f32, s_g = sum_q (b_g - 0.5 T)*Y
//   (16-lane butterfly reduce), streaming logsumexp over groups.
// ---------------------------------------------------------------------------
__global__ __launch_bounds__(256) void gmm_main_kernel(
    const float* __restrict__ X,
    const float* __restrict__ lambdas,
    const float* __restrict__ ws,
    float* __restrict__ tileOut,
    int nTiles) {

    __shared__ float lds[8][16 * 18];  // 8 waves per block, stride-18 rows

    const float* Sinv = ws + WS_SINV;
    const float* bvec = ws + WS_BVEC;
    const float* Kc   = ws + WS_KC;

    const int lane = threadIdx.x & 31;
    const int wv   = threadIdx.x >> 5;
    const int half = lane >> 4;       // 0: lanes 0-15, 1: lanes 16-31
    const int lr   = lane & 15;
    const int c0   = 8 * half;        // this lane's column window for transforms
    float* tile = &lds[wv][0];

    // Hoisted wave-invariant operands -------------------------------------
    float lamv[8], il[8], lm1[8];
#pragma unroll
    for (int i = 0; i < 8; ++i) {
        float lv = lambdas[c0 + i];
        lamv[i] = lv; il[i] = 1.f / lv; lm1[i] = lv - 1.f;
    }
    // B operand (ISA B-layout, mirrors A): slice j, VGPR0 = row 4j+2*half,
    // VGPR1 = row 4j+1+2*half, column = lane%16
    v2f Bop[G8][4];
#pragma unroll
    for (int g = 0; g < G8; ++g)
#pragma unroll
        for (int j = 0; j < 4; ++j) {
            Bop[g][j].x = Sinv[g * 256 + (4 * j + 2 * half + 0) * 16 + lr];
            Bop[g][j].y = Sinv[g * 256 + (4 * j + 2 * half + 1) * 16 + lr];
        }
    float bb[G8], Kg[G8];
#pragma unroll
    for (int g = 0; g < G8; ++g) { bb[g] = bvec[g * 16 + lr]; Kg[g] = Kc[g]; }

    const int waveId = blockIdx.x * (blockDim.x >> 5) + wv;
    const int nWaves = gridDim.x * (blockDim.x >> 5);

    for (int t = waveId; t < nTiles; t += nWaves) {
        // --- load X rows, Box-Cox + jacobian, stage Xt in LDS ------------
        const float4* xp =
            (const float4*)(X + ((size_t)(t * 16 + lr)) * 16 + c0);
        float4 x0 = xp[0], x1 = xp[1];
        float xs[8] = {x0.x, x0.y, x0.z, x0.w, x1.x, x1.y, x1.z, x1.w};
        float jp = 0.f;
#pragma unroll
        for (int i = 0; i < 8; ++i) {
            float lx = __logf(xs[i]);
            float xt = (__expf(lamv[i] * lx) - 1.f) * il[i];
            jp += lm1[i] * lx;
            tile[lr * 18 + c0 + i] = xt;
        }
        __builtin_amdgcn_wave_barrier();  // wave-local LDS; DS ops are in-order

        // --- A-layout (16x4 f32 slices) and C-layout reads ---------------
        v2f a[4];
#pragma unroll
        for (int j = 0; j < 4; ++j) {
            a[j].x = tile[lr * 18 + 4 * j + 2 * half + 0];
            a[j].y = tile[lr * 18 + 4 * j + 2 * half + 1];
        }
        float yc[8];
#pragma unroll
        for (int r = 0; r < 8; ++r) yc[r] = tile[(r + 8 * half) * 18 + lr];

        // jacobian: per-row sum, then sum over the 16 rows of the tile
        float jfull = jp + __shfl_xor(jp, 16, 32);
        float jacsum = jfull;
#pragma unroll
        for (int s = 1; s <= 8; s <<= 1) jacsum += __shfl_xor(jacsum, s, 32);

        // --- per-group quadratic form via f32 WMMA + streaming LSE -------
        float mx[8], sm[8];
#pragma unroll
        for (int g = 0; g < G8; ++g) {
            v8f acc = {0.f, 0.f, 0.f, 0.f, 0.f, 0.f, 0.f, 0.f};
#pragma unroll
            for (int j = 0; j < 4; ++j)
                acc = __builtin_amdgcn_wmma_f32_16x16x4_f32(
                    false, a[j], false, Bop[g][j], (short)0, acc, false, false);
#pragma unroll
            for (int r = 0; r < 8; ++r) {
                // s contribution: (b_q - 0.5*(Y B)_{m,q}) * Y_{m,q}, sum over q
                float v = (bb[g] - 0.5f * acc[r]) * yc[r];
#pragma unroll
                for (int s = 1; s <= 8; s <<= 1) v += __shfl_xor(v, s, 32);
                float val = v + Kg[g];
                if (g == 0) { mx[r] = val; sm[r] = 1.f; }
                else {
                    float m2 = fmaxf(mx[r], val);
                    sm[r] = sm[r] * __expf(mx[r] - m2) + __expf(val - m2);
                    mx[r] = m2;
                }
            }
        }
        // lse per point; points r (lanes 0-15) and r+8 (lanes 16-31)
        float t2 = 0.f;
#pragma unroll
        for (int r = 0; r < 8; ++r) t2 += mx[r] + __logf(sm[r]);
        float cross = t2 + __shfl_xor(t2, 16, 32);  // all 16 points
        if (lane == 0)
            tileOut[t] = -(cross + jacsum + 16.f * LEAD_F);
        __builtin_amdgcn_wave_barrier();
    }
}

// ---------------------------------------------------------------------------
// Deterministic fixed-order reduction of tile sums -> scalar NLL
// ---------------------------------------------------------------------------
__global__ void gmm_reduce_kernel(const float* __restrict__ tileOut, int n,
                                  float* __restrict__ out) {
    __shared__ float sh[256];
    float s = 0.f;
    for (int i = threadIdx.x; i < n; i += 256) s += tileOut[i];
    sh[threadIdx.x] = s;
    __syncthreads();
    for (int off = 128; off > 0; off >>= 1) {
        if ((int)threadIdx.x < off) sh[threadIdx.x] += sh[threadIdx.x + off];
        __syncthreads();
    }
    if (threadIdx.x == 0) out[0] = sh[0];
}

extern "C" void kernel_launch(void* const* d_in, const int* in_sizes, int n_in,
                              void* d_out, int out_size, void* d_ws, size_t ws_size,
                              hipStream_t stream) {
    const float* X       = (const float*)d_in[0];
    const float* lambdas = (const float*)d_in[1];
    const float* mus     = (const float*)d_in[2];
    const float* sigmas  = (const float*)d_in[3];
    const float* w       = (const float*)d_in[4];
    float* ws = (float*)d_ws;

    const int N = in_sizes[0] / 16;
    const int nTiles = N / 16;

    gmm_prep_kernel<<<1, 32, 0, stream>>>(mus, sigmas, w, ws);
    gmm_main_kernel<<<1024, 256, 0, stream>>>(X, lambdas, ws, ws + WS_TILES,
                                              nTiles);
    gmm_reduce_kernel<<<1, 256, 0, stream>>>(ws + WS_TILES, nTiles,
                                             (float*)d_out);
}